// riemannian_exp_4930622456055
// MI455X (gfx1250) — compile-verified
//
#include <hip/hip_runtime.h>
#include <math.h>

// ---------------------------------------------------------------------------
// Pose bin-classification + Rodrigues rotation composition (MI455X / gfx1250)
//   out[b] = key_poses[argmax(ybin[b,:])] @ exp(hat(yres[b]))
// Memory-bound: 448 MB traffic -> ~19 us floor @ 23.3 TB/s. One thread / row.
// ---------------------------------------------------------------------------

typedef float v4f __attribute__((ext_vector_type(4)));
typedef int   v4i __attribute__((ext_vector_type(4)));

// Address-space-qualified 16-byte vector types for the async-DMA builtin.
typedef __attribute__((address_space(1))) v4i gv4i;   // global
typedef __attribute__((address_space(3))) v4i lv4i;   // LDS

#define NCLS 100          // C
#define TPB  256          // 8 wave32 per block

// ---- CDNA5 async global->LDS copy (16 B per active lane) -------------------
__device__ __forceinline__ void async_copy_b128(const float* g, float* l) {
#if __has_builtin(__builtin_amdgcn_global_load_async_to_lds_b128)
    __builtin_amdgcn_global_load_async_to_lds_b128(
        (gv4i*)(void*)(const_cast<float*>(g)),
        (lv4i*)(void*)l,
        /*offset=*/0, /*cpol=*/0);
#else
    // vdst = LDS byte offset VGPR, vaddr = 64-bit global address VGPR pair
    unsigned lds_off =
        (unsigned)(unsigned long long)(__attribute__((address_space(3))) void*)l;
    asm volatile("global_load_async_to_lds_b128 %0, %1, off"
                 :: "v"(lds_off), "v"(g)
                 : "memory");
#endif
}

__device__ __forceinline__ void wait_async0() {
#if __has_builtin(__builtin_amdgcn_s_wait_asynccnt)
    __builtin_amdgcn_s_wait_asynccnt(0);
#else
    asm volatile("s_wait_asynccnt 0" ::: "memory");
#endif
}

__global__ __launch_bounds__(TPB)
void pose_compose_kernel(const float* __restrict__ ybin,      // [B,100]
                         const float* __restrict__ yres,      // [B,3]
                         const float* __restrict__ key_poses, // [100,3,3]
                         float* __restrict__ out,             // [B,3,3]
                         int B)
{
    // Stage the 900-float (3.6 KB) key-pose table into LDS with the CDNA5
    // async DMA path: 225 lanes x 16 B.
    __shared__ float skp[NCLS * 9];
    const int t = threadIdx.x;
    if (t < (NCLS * 9) / 4) {                 // 225 chunks of float4
        async_copy_b128(key_poses + t * 4, &skp[t * 4]);
    }
    wait_async0();
    __syncthreads();

    const int r = blockIdx.x * TPB + t;
    if (r >= B) return;

    // ---- argmax over 100 logits: 25 x non-temporal b128 loads -------------
    const v4f* row = (const v4f*)(ybin + (size_t)r * NCLS);
    float best = -INFINITY;
    int   bi   = 0;
#pragma unroll
    for (int j = 0; j < NCLS / 4; ++j) {
        v4f v = __builtin_nontemporal_load(&row[j]);
        const int base = j * 4;
        if (v[0] > best) { best = v[0]; bi = base + 0; }
        if (v[1] > best) { best = v[1]; bi = base + 1; }
        if (v[2] > best) { best = v[2]; bi = base + 2; }
        if (v[3] > best) { best = v[3]; bi = base + 3; }
    }

    // ---- Rodrigues: R = I + sin(t)K + (1-cos(t))K^2 -----------------------
    const float* yr = yres + (size_t)r * 3;
    const float x = yr[0], y = yr[1], z = yr[2];
    const float n2    = fmaf(x, x, fmaf(y, y, z * z));
    const float angle = sqrtf(n2);
    const float inv   = 1.0f / fmaxf(angle, 1e-12f);
    const float ax = x * inv, ay = y * inv, az = z * inv;
    float s, cc;
    sincosf(angle, &s, &cc);
    const float c = 1.0f - cc;
    const float nn = fmaf(ax, ax, fmaf(ay, ay, az * az)); // ~1; K^2 = aa^T - (a.a)I exactly

    // R[i][j] = d_ij + s*K[i][j] + c*(a_i a_j - d_ij*nn)
    const float r00 = fmaf(c, fmaf(ax, ax, -nn), 1.0f);
    const float r01 = fmaf(c, ax * ay, -s * az);
    const float r02 = fmaf(c, ax * az,  s * ay);
    const float r10 = fmaf(c, ay * ax,  s * az);
    const float r11 = fmaf(c, fmaf(ay, ay, -nn), 1.0f);
    const float r12 = fmaf(c, ay * az, -s * ax);
    const float r20 = fmaf(c, az * ax, -s * ay);
    const float r21 = fmaf(c, az * ay,  s * ax);
    const float r22 = fmaf(c, fmaf(az, az, -nn), 1.0f);

    // ---- gather selected key pose from LDS, O = P @ R ---------------------
    const float* P = &skp[bi * 9];
    const float p00 = P[0], p01 = P[1], p02 = P[2];
    const float p10 = P[3], p11 = P[4], p12 = P[5];
    const float p20 = P[6], p21 = P[7], p22 = P[8];

    float* o = out + (size_t)r * 9;
    __builtin_nontemporal_store(fmaf(p00, r00, fmaf(p01, r10, p02 * r20)), &o[0]);
    __builtin_nontemporal_store(fmaf(p00, r01, fmaf(p01, r11, p02 * r21)), &o[1]);
    __builtin_nontemporal_store(fmaf(p00, r02, fmaf(p01, r12, p02 * r22)), &o[2]);
    __builtin_nontemporal_store(fmaf(p10, r00, fmaf(p11, r10, p12 * r20)), &o[3]);
    __builtin_nontemporal_store(fmaf(p10, r01, fmaf(p11, r11, p12 * r21)), &o[4]);
    __builtin_nontemporal_store(fmaf(p10, r02, fmaf(p11, r12, p12 * r22)), &o[5]);
    __builtin_nontemporal_store(fmaf(p20, r00, fmaf(p21, r10, p22 * r20)), &o[6]);
    __builtin_nontemporal_store(fmaf(p20, r01, fmaf(p21, r11, p22 * r21)), &o[7]);
    __builtin_nontemporal_store(fmaf(p20, r02, fmaf(p21, r12, p22 * r22)), &o[8]);
}

extern "C" void kernel_launch(void* const* d_in, const int* in_sizes, int n_in,
                              void* d_out, int out_size, void* d_ws, size_t ws_size,
                              hipStream_t stream) {
    const float* ybin      = (const float*)d_in[0]; // [B,100] f32
    const float* yres      = (const float*)d_in[1]; // [B,3]   f32
    const float* key_poses = (const float*)d_in[2]; // [100,3,3] f32
    float*       out       = (float*)d_out;         // [B,3,3] f32

    const int B = in_sizes[1] / 3;
    if (B <= 0) return;
    const int blocks = (B + TPB - 1) / TPB;
    hipLaunchKernelGGL(pose_compose_kernel, dim3(blocks), dim3(TPB), 0, stream,
                       ybin, yres, key_poses, out, B);
}